// GAT_18940805775884
// MI455X (gfx1250) — compile-verified
//
#include <hip/hip_runtime.h>
#include <hip/hip_bf16.h>

typedef __attribute__((ext_vector_type(2))) float v2f;
typedef __attribute__((ext_vector_type(8))) float v8f;

#define NEG_SLOPE 0.2f

__device__ __forceinline__ float leaky(float x) { return x > 0.f ? x : NEG_SLOPE * x; }

// Float atomic max via sign-aware integer atomics (native int atomics, no CAS loop).
__device__ __forceinline__ void atomicMaxF(float* a, float v) {
  if (v >= 0.f) atomicMax((int*)a, __float_as_int(v));
  else          atomicMin((unsigned int*)a, __float_as_uint(v));
}

// ---------------------------------------------------------------------------
// H[nrows x NCOLS] = X[nrows x 128] * W[128 x NCOLS]  (fp32-exact, WMMA f32)
// 256 threads = 8 waves; each wave -> one 16x16 tile of a 32x64 block.
// ---------------------------------------------------------------------------
template<int NCOLS>
__global__ void __launch_bounds__(256)
gemm_wmma_f32(const float* __restrict__ X, const float* __restrict__ W,
              float* __restrict__ H, int nrows) {
  constexpr int SW = 80;   // 64 + pad: 2*SW mod 64 == 32 -> half-waves on disjoint banks
  constexpr int SX = 132;  // 128 + pad: banks {4m} vs {4m+2} -> conflict-free
  __shared__ float sW[128 * SW];
  __shared__ float sX[32 * SX];
  const int tid = threadIdx.x;
  const int bx = blockIdx.x, by = blockIdx.y;

  // Stage the 128x64 W slab for this column block (coalesced).
  for (int i = tid; i < 128 * 64; i += 256) {
    const int r = i >> 6, c = i & 63;
    sW[r * SW + c] = W[r * NCOLS + by * 64 + c];
  }
  // Stage the 32x128 X tile (zero-pad past nrows).
  for (int i = tid; i < 32 * 128; i += 256) {
    const int r = i >> 7, c = i & 127;
    const int row = bx * 32 + r;
    sX[r * SX + c] = (row < nrows) ? X[(size_t)row * 128 + c] : 0.f;
  }
  __syncthreads();

  const int wave = tid >> 5, lane = tid & 31;
  const int rt = wave >> 2, ct = wave & 3;      // 2 row-tiles x 4 col-tiles
  const int m    = lane & 15;                   // M (A) / N (B,C,D) index
  const int koff = (lane >> 4) << 1;            // lanes 16-31 carry K+2,K+3

  v8f acc = {};
  const float* xrow = &sX[(rt * 16 + m) * SX];
  const float* wcol = &sW[ct * 16 + m];
  #pragma unroll 8
  for (int k = 0; k < 128; k += 4) {
    v2f a, b;
    a.x = xrow[k + koff];
    a.y = xrow[k + koff + 1];
    b.x = wcol[(k + koff) * SW];
    b.y = wcol[(k + koff + 1) * SW];
    // D = A(16x4,f32) x B(4x16,f32) + C  -> v_wmma_f32_16x16x4_f32
    acc = __builtin_amdgcn_wmma_f32_16x16x4_f32(false, a, false, b,
                                                (short)0, acc, false, false);
  }

  const int half = lane >> 4;                   // C/D: M = 8*half + vgpr, N = lane&15
  #pragma unroll
  for (int j = 0; j < 8; ++j) {
    const int row = bx * 32 + rt * 16 + half * 8 + j;
    if (row < nrows)
      H[(size_t)row * NCOLS + by * 64 + ct * 16 + m] = acc[j];
  }
}

// ---------------------------------------------------------------------------
// Init: zero aggregation buffers & output, -inf maxes, zero sums.
// ---------------------------------------------------------------------------
__global__ void init_kernel(float* __restrict__ agg1, float* __restrict__ m1,
                            float* __restrict__ s1, float* __restrict__ m2,
                            float* __restrict__ s2, float* __restrict__ out, int N) {
  const int i = blockIdx.x * blockDim.x + threadIdx.x;
  const float ninf = -__builtin_inff();
  if (i < N * 128) agg1[i] = 0.f;
  if (i < N * 64)  out[i]  = 0.f;
  if (i < N * 4)   { m1[i] = ninf; s1[i] = 0.f; }
  if (i < N)       { m2[i] = ninf; s2[i] = 0.f; }
}

// ---------------------------------------------------------------------------
// Per-node attention coefficients.
// ---------------------------------------------------------------------------
__global__ void alpha1_kernel(const float* __restrict__ h1,
                              const float* __restrict__ a_src,
                              const float* __restrict__ a_dst,
                              float* __restrict__ as1, float* __restrict__ ad1, int N) {
  const int t = blockIdx.x * blockDim.x + threadIdx.x;
  if (t >= N * 4) return;
  const int n = t >> 2, h = t & 3;
  const float* row = h1 + (size_t)n * 128 + h * 32;
  float s = 0.f, d = 0.f;
  #pragma unroll
  for (int c = 0; c < 32; ++c) {
    const float v = row[c];
    s += v * a_src[h * 32 + c];
    d += v * a_dst[h * 32 + c];
  }
  as1[t] = s; ad1[t] = d;
}

__global__ void alpha2_kernel(const float* __restrict__ h2,
                              const float* __restrict__ a_src,
                              const float* __restrict__ a_dst,
                              float* __restrict__ as2, float* __restrict__ ad2, int N) {
  const int n = blockIdx.x * blockDim.x + threadIdx.x;
  if (n >= N) return;
  const float* row = h2 + (size_t)n * 64;
  float s = 0.f, d = 0.f;
  #pragma unroll
  for (int c = 0; c < 64; ++c) {
    const float v = row[c];
    s += v * a_src[c];
    d += v * a_dst[c];
  }
  as2[n] = s; ad2[n] = d;
}

// ---------------------------------------------------------------------------
// Segment-softmax over dst: pass 1 (max), pass 2 (sum of exp).
// Edge ids >= E are the synthesized self-loops.
// ---------------------------------------------------------------------------
template<int H>
__global__ void edge_max_kernel(const int* __restrict__ ei,
                                const float* __restrict__ as, const float* __restrict__ ad,
                                float* __restrict__ m, int E, int Etot) {
  const int e = blockIdx.x * blockDim.x + threadIdx.x;
  if (e >= Etot) return;
  int src, dst;
  if (e < E) { src = ei[e]; dst = ei[E + e]; } else { src = dst = e - E; }
  #pragma unroll
  for (int h = 0; h < H; ++h) {
    const float l = leaky(as[src * H + h] + ad[dst * H + h]);
    atomicMaxF(&m[dst * H + h], l);
  }
}

template<int H>
__global__ void edge_sum_kernel(const int* __restrict__ ei,
                                const float* __restrict__ as, const float* __restrict__ ad,
                                const float* __restrict__ m, float* __restrict__ s,
                                int E, int Etot) {
  const int e = blockIdx.x * blockDim.x + threadIdx.x;
  if (e >= Etot) return;
  int src, dst;
  if (e < E) { src = ei[e]; dst = ei[E + e]; } else { src = dst = e - E; }
  #pragma unroll
  for (int h = 0; h < H; ++h) {
    const float l = leaky(as[src * H + h] + ad[dst * H + h]);
    atomicAdd(&s[dst * H + h], expf(l - m[dst * H + h]));
  }
}

// ---------------------------------------------------------------------------
// Attention-weighted message scatter: one wave per edge (coalesced 128B
// gather of h[src] + coalesced atomic adds into agg[dst], L2-resident).
// ---------------------------------------------------------------------------
__global__ void edge_scatter1_kernel(const int* __restrict__ ei,
                                     const float* __restrict__ h1,
                                     const float* __restrict__ as, const float* __restrict__ ad,
                                     const float* __restrict__ m, const float* __restrict__ s,
                                     float* __restrict__ agg, int E, int Etot) {
  const long t = (long)blockIdx.x * blockDim.x + threadIdx.x;
  const int e = (int)(t >> 5);
  const int lane = (int)(t & 31);
  if (e >= Etot) return;
  int src, dst;
  if (e < E) { src = ei[e]; dst = ei[E + e]; } else { src = dst = e - E; }
  #pragma unroll
  for (int h = 0; h < 4; ++h) {
    const float l = leaky(as[src * 4 + h] + ad[dst * 4 + h]);
    const float alpha = expf(l - m[dst * 4 + h]) / (s[dst * 4 + h] + 1e-16f);
    const int c = h * 32 + lane;
    atomicAdd(&agg[(size_t)dst * 128 + c], h1[(size_t)src * 128 + c] * alpha);
  }
}

__global__ void edge_scatter2_kernel(const int* __restrict__ ei,
                                     const float* __restrict__ h2,
                                     const float* __restrict__ as, const float* __restrict__ ad,
                                     const float* __restrict__ m, const float* __restrict__ s,
                                     float* __restrict__ outp, int E, int Etot) {
  const long t = (long)blockIdx.x * blockDim.x + threadIdx.x;
  const int e = (int)(t >> 5);
  const int lane = (int)(t & 31);
  if (e >= Etot) return;
  int src, dst;
  if (e < E) { src = ei[e]; dst = ei[E + e]; } else { src = dst = e - E; }
  const float l = leaky(as[src] + ad[dst]);
  const float alpha = expf(l - m[dst]) / (s[dst] + 1e-16f);
  atomicAdd(&outp[(size_t)dst * 64 + lane],      h2[(size_t)src * 64 + lane] * alpha);
  atomicAdd(&outp[(size_t)dst * 64 + 32 + lane], h2[(size_t)src * 64 + 32 + lane] * alpha);
}

// ---------------------------------------------------------------------------
// Finalizers: bias (+ReLU for layer 1, in place -> layer-2 GEMM input).
// ---------------------------------------------------------------------------
__global__ void finalize1_kernel(float* __restrict__ agg1, const float* __restrict__ b1, int N) {
  const int i = blockIdx.x * blockDim.x + threadIdx.x;
  if (i >= N * 128) return;
  const float v = agg1[i] + b1[i & 127];
  agg1[i] = v > 0.f ? v : 0.f;
}

__global__ void finalize2_kernel(float* __restrict__ out, const float* __restrict__ b2, int N) {
  const int i = blockIdx.x * blockDim.x + threadIdx.x;
  if (i >= N * 64) return;
  out[i] = out[i] + b2[i & 63];
}

// ---------------------------------------------------------------------------
extern "C" void kernel_launch(void* const* d_in, const int* in_sizes, int n_in,
                              void* d_out, int out_size, void* d_ws, size_t ws_size,
                              hipStream_t stream) {
  const float* x      = (const float*)d_in[0];
  const int*   ei     = (const int*)d_in[1];   // edge_index flat [2, E]
  const float* W1     = (const float*)d_in[2];
  const float* a_src1 = (const float*)d_in[3];
  const float* a_dst1 = (const float*)d_in[4];
  const float* b1     = (const float*)d_in[5];
  const float* W2     = (const float*)d_in[6];
  const float* a_src2 = (const float*)d_in[7];
  const float* a_dst2 = (const float*)d_in[8];
  const float* b2     = (const float*)d_in[9];
  float* out = (float*)d_out;

  const int N = in_sizes[0] / 128;
  const int E = in_sizes[1] / 2;
  const int Etot = E + N;   // + self loops

  // Workspace carve-up (~68 MB).
  float* w    = (float*)d_ws;
  float* h1   = w; w += (size_t)N * 128;
  float* agg1 = w; w += (size_t)N * 128;   // becomes relu(agg1+b1) in place
  float* as1  = w; w += (size_t)N * 4;
  float* ad1  = w; w += (size_t)N * 4;
  float* m1   = w; w += (size_t)N * 4;
  float* s1   = w; w += (size_t)N * 4;
  float* h2   = w; w += (size_t)N * 64;
  float* as2  = w; w += N;
  float* ad2  = w; w += N;
  float* m2   = w; w += N;
  float* s2   = w; w += N;

  const int T = 256;
  const int gN128 = (N * 128 + T - 1) / T;
  const int gN64  = (N * 64 + T - 1) / T;
  const int gE    = (Etot + T - 1) / T;
  const int gEw   = (int)(((long)Etot * 32 + T - 1) / T);  // wave-per-edge grids

  init_kernel<<<gN128, T, 0, stream>>>(agg1, m1, s1, m2, s2, out, N);

  // ---- Layer 1 ----
  dim3 g1((N + 31) / 32, 2);
  gemm_wmma_f32<128><<<g1, 256, 0, stream>>>(x, W1, h1, N);
  alpha1_kernel<<<(N * 4 + T - 1) / T, T, 0, stream>>>(h1, a_src1, a_dst1, as1, ad1, N);
  edge_max_kernel<4><<<gE, T, 0, stream>>>(ei, as1, ad1, m1, E, Etot);
  edge_sum_kernel<4><<<gE, T, 0, stream>>>(ei, as1, ad1, m1, s1, E, Etot);
  edge_scatter1_kernel<<<gEw, T, 0, stream>>>(ei, h1, as1, ad1, m1, s1, agg1, E, Etot);
  finalize1_kernel<<<gN128, T, 0, stream>>>(agg1, b1, N);

  // ---- Layer 2 ----
  dim3 g2((N + 31) / 32, 1);
  gemm_wmma_f32<64><<<g2, 256, 0, stream>>>(agg1, W2, h2, N);
  alpha2_kernel<<<(N + T - 1) / T, T, 0, stream>>>(h2, a_src2, a_dst2, as2, ad2, N);
  edge_max_kernel<1><<<gE, T, 0, stream>>>(ei, as2, ad2, m2, E, Etot);
  edge_sum_kernel<1><<<gE, T, 0, stream>>>(ei, as2, ad2, m2, s2, E, Etot);
  edge_scatter2_kernel<<<gEw, T, 0, stream>>>(ei, h2, as2, ad2, m2, s2, out, E, Etot);
  finalize2_kernel<<<gN64, T, 0, stream>>>(out, b2, N);
}